// Model_24507083391146
// MI455X (gfx1250) — compile-verified
//
#include <hip/hip_runtime.h>
#include <math.h>

#define NTYPES 2
#define NNODES 50000
#define NEDGES 400000
#define CDIM 128
#define HDIM 4
#define DDIM 32
#define NLAYERS 2

typedef __attribute__((ext_vector_type(2))) float v2f;
typedef __attribute__((ext_vector_type(8))) float v8f;

__device__ __forceinline__ float gelu_tanh(float x) {
    const float c = 0.7978845608028654f; // sqrt(2/pi)
    float x3 = x * x * x;
    return 0.5f * x * (1.0f + tanhf(c * (x + 0.044715f * x3)));
}

// float atomic max via monotonic int encoding (works for mixed signs)
__device__ __forceinline__ void atomicMaxF(float* addr, float val) {
    if (val >= 0.0f) atomicMax((int*)addr, __float_as_int(val));
    else             atomicMin((unsigned int*)addr, __float_as_uint(val));
}

// ---------------------------------------------------------------------------
// Y[nRows,128] = (applyGelu? gelu(X) : X)[nRows,128] @ W[128,128] + bias[128]
// One wave -> 16 rows x 128 cols via v_wmma_f32_16x16x4_f32.
// W is staged into LDS in two 64-K-row phases, PRE-SWIZZLED into the ISA
// B-fragment layout: sW[kt][nt][lane] = float2{ B[k0+e0][n], B[k0+e0+1][n] }
// with lanes 0-15 holding K={0,1} and lanes 16-31 holding K={2,3}. Each
// fragment is then ONE conflict-free ds_load_b64 (32 lanes x 8B = 64 banks).
// ---------------------------------------------------------------------------
__global__ __launch_bounds__(256, 2) void gemm128_bias(
    const float* __restrict__ X, const float* __restrict__ W,
    const float* __restrict__ bias, float* __restrict__ Y,
    int nRows, int applyGelu)
{
    __shared__ float2 sW[16 * 8 * 32];   // 32 KB: 16 kt x 8 nt x 32 lanes
    __shared__ float  sB[CDIM];

    const int tid = threadIdx.x;
    if (tid < CDIM) sB[tid] = bias[tid];

    const int wave     = tid >> 5;
    const int lane     = tid & 31;
    const int laneHalf = lane & 15;
    const int hi       = lane >> 4;      // 0: K={0,1}, 1: K={2,3} per ISA A layout
    const int m0       = blockIdx.x * 128 + wave * 16;

    int rowA = m0 + laneHalf;
    if (rowA >= nRows) rowA = nRows - 1;           // clamp; tail masked on store
    const float* __restrict__ Xrow = X + (size_t)rowA * CDIM;

    v8f acc[8] = {};

    for (int phase = 0; phase < 2; ++phase) {
        __syncthreads();
        // stage 64 K-rows of W (8192 floats) swizzled into fragment layout
        for (int i = tid; i < 2048; i += 256) {
            const int r  = i >> 5;               // local k row: 0..63
            const int c  = (i & 31) * 4;         // col 0..124 step 4
            const float4 w4 =
                *(const float4*)(W + (size_t)(phase * 64 + r) * CDIM + c);
            const int ktl      = r >> 2;         // local kt 0..15
            const int kk       = r & 3;
            const int elem     = kk & 1;         // slot within float2
            const int laneBase = (kk >> 1) * 16; // 0 or 16
            const int nt       = c >> 4;
            const int nl       = c & 15;
            const float vals[4] = { w4.x, w4.y, w4.z, w4.w };
#pragma unroll
            for (int j = 0; j < 4; ++j) {
                float* dst = (float*)&sW[(ktl * 8 + nt) * 32 + laneBase + nl + j];
                dst[elem] = vals[j];
            }
            if (phase == 0) {   // warm L2/WGP$ for the second half of W
                __builtin_prefetch(W + (size_t)(64 + r) * CDIM + c, 0, 0);
            }
        }
        __syncthreads();

        for (int kt = 0; kt < 16; ++kt) {
            const int kg = phase * 64 + kt * 4 + hi * 2;   // global K for A frag
            v2f a;
            float a0 = Xrow[kg];
            float a1 = Xrow[kg + 1];
            if (applyGelu) { a0 = gelu_tanh(a0); a1 = gelu_tanh(a1); }
            a.x = a0; a.y = a1;
            const float2* __restrict__ bBase = &sW[kt * 8 * 32 + lane];
#pragma unroll
            for (int nt = 0; nt < 8; ++nt) {
                const float2 bf = bBase[nt * 32];   // one ds_load_b64, no movs
                v2f b; b.x = bf.x; b.y = bf.y;
                acc[nt] = __builtin_amdgcn_wmma_f32_16x16x4_f32(
                    false, a, false, b, (short)0, acc[nt], false, false);
            }
        }
    }

#pragma unroll
    for (int nt = 0; nt < 8; ++nt) {
        const int col = nt * 16 + laneHalf;
        const float bi = sB[col];
#pragma unroll
        for (int r = 0; r < 8; ++r) {
            const int row = m0 + hi * 8 + r;       // ISA C/D layout
            if (row < nRows) Y[(size_t)row * CDIM + col] = acc[nt][r] + bi;
        }
    }
}

// ---------------------------------------------------------------------------
// Edge pass 1: logits[e,h] = (k[si] @ a_rel[h]) . q[di] * p[h] * scale
// and segment-max into amax[di,h]. One wave per edge, loop heads.
// ---------------------------------------------------------------------------
__global__ __launch_bounds__(256) void edge_logits(
    const float* __restrict__ kmat, const float* __restrict__ qmat,
    const int* __restrict__ esrc, const int* __restrict__ edst,
    const float* __restrict__ arel, const float* __restrict__ prel,
    float* __restrict__ logit, float* __restrict__ amax, int nEdges)
{
    __shared__ float sA[HDIM * DDIM * DDIM];   // 16 KB
    __shared__ float sP[HDIM];
    const int tid = threadIdx.x;
    for (int i = tid; i < HDIM * DDIM * DDIM; i += 256) sA[i] = arel[i];
    if (tid < HDIM) sP[tid] = prel[tid];
    __syncthreads();

    const int wave = tid >> 5, lane = tid & 31;
    const int e = blockIdx.x * 8 + wave;
    if (e >= nEdges) return;
    const int si = esrc[e], di = edst[e];
    const float scale = 0.17677669529663687f;  // 1/sqrt(32)

#pragma unroll
    for (int h = 0; h < HDIM; ++h) {
        const float kv = kmat[(size_t)si * CDIM + h * DDIM + lane];
        const float qv = qmat[(size_t)di * CDIM + h * DDIM + lane];
        float krel = 0.0f;
#pragma unroll
        for (int d = 0; d < DDIM; ++d)
            krel = fmaf(__shfl(kv, d, 32), sA[(h * DDIM + d) * DDIM + lane], krel);
        float part = qv * krel;
#pragma unroll
        for (int off = 16; off > 0; off >>= 1) part += __shfl_xor(part, off, 32);
        if (lane == 0) {
            const float val = part * sP[h] * scale;
            logit[(size_t)e * HDIM + h] = val;
            atomicMaxF(&amax[(size_t)di * HDIM + h], val);
        }
    }
}

// ---------------------------------------------------------------------------
// Edge pass 2: w = exp(logit - amax[di]); den[di] += w  (in-place on logit buf)
// ---------------------------------------------------------------------------
__global__ __launch_bounds__(256) void edge_exp(
    float* __restrict__ wlog, const int* __restrict__ edst,
    const float* __restrict__ amax, float* __restrict__ den, int nEdges)
{
    const int i = blockIdx.x * blockDim.x + threadIdx.x;
    if (i >= nEdges * HDIM) return;
    const int e = i >> 2, h = i & 3;
    const int di = edst[e];
    const float val = expf(wlog[i] - amax[(size_t)di * HDIM + h]);
    wlog[i] = val;
    atomicAdd(&den[(size_t)di * HDIM + h], val);
}

// ---------------------------------------------------------------------------
// Edge pass 3: out[di] += (w/den[di]) * (v[si] @ m_rel[h]). Wave per edge.
// ---------------------------------------------------------------------------
__global__ __launch_bounds__(256) void edge_agg(
    const float* __restrict__ vmat, const int* __restrict__ esrc,
    const int* __restrict__ edst, const float* __restrict__ mrel,
    const float* __restrict__ wlog, const float* __restrict__ den,
    float* __restrict__ outagg, int nEdges)
{
    __shared__ float sM[HDIM * DDIM * DDIM];
    const int tid = threadIdx.x;
    for (int i = tid; i < HDIM * DDIM * DDIM; i += 256) sM[i] = mrel[i];
    __syncthreads();

    const int wave = tid >> 5, lane = tid & 31;
    const int e = blockIdx.x * 8 + wave;
    if (e >= nEdges) return;
    const int si = esrc[e], di = edst[e];

#pragma unroll
    for (int h = 0; h < HDIM; ++h) {
        const float vv = vmat[(size_t)si * CDIM + h * DDIM + lane];
        const float coef = wlog[(size_t)e * HDIM + h] /
                           (den[(size_t)di * HDIM + h] + 1e-16f);
        float vrel = 0.0f;
#pragma unroll
        for (int d = 0; d < DDIM; ++d)
            vrel = fmaf(__shfl(vv, d, 32), sM[(h * DDIM + d) * DDIM + lane], vrel);
        atomicAdd(&outagg[(size_t)di * CDIM + h * DDIM + lane], coef * vrel);
    }
}

// ---------------------------------------------------------------------------
// x' = relu(LN(beta*o + (1-beta)*x) * g + b). One wave per row (128 feats).
// ---------------------------------------------------------------------------
__global__ __launch_bounds__(256) void skip_ln_relu(
    const float* __restrict__ o, const float* __restrict__ xin,
    const float* __restrict__ skipv, const float* __restrict__ lng,
    const float* __restrict__ lnb, float* __restrict__ xout, int rowsTotal)
{
    const int row = blockIdx.x * 8 + (threadIdx.x >> 5);
    if (row >= rowsTotal) return;
    const int lane = threadIdx.x & 31;
    const int t = row / NNODES;
    const float beta = 1.0f / (1.0f + expf(-skipv[t]));

    float vals[4];
    float s = 0.0f, s2 = 0.0f;
#pragma unroll
    for (int j = 0; j < 4; ++j) {
        const int f = lane + 32 * j;
        const float val = beta * o[(size_t)row * CDIM + f] +
                          (1.0f - beta) * xin[(size_t)row * CDIM + f];
        vals[j] = val; s += val; s2 += val * val;
    }
#pragma unroll
    for (int off = 16; off > 0; off >>= 1) {
        s  += __shfl_xor(s, off, 32);
        s2 += __shfl_xor(s2, off, 32);
    }
    const float mu  = s * (1.0f / CDIM);
    const float var = s2 * (1.0f / CDIM) - mu * mu;
    const float inv = rsqrtf(var + 1e-5f);
#pragma unroll
    for (int j = 0; j < 4; ++j) {
        const int f = lane + 32 * j;
        const float y = (vals[j] - mu) * inv * lng[t * CDIM + f] + lnb[t * CDIM + f];
        xout[(size_t)row * CDIM + f] = fmaxf(y, 0.0f);
    }
}

__global__ void fill_f32(float* __restrict__ p, float v, size_t n) {
    const size_t i = (size_t)blockIdx.x * blockDim.x + threadIdx.x;
    if (i < n) p[i] = v;
}

// ---------------------------------------------------------------------------
extern "C" void kernel_launch(void* const* d_in, const int* in_sizes, int n_in,
                              void* d_out, int out_size, void* d_ws, size_t ws_size,
                              hipStream_t stream)
{
    const float* x0   = (const float*)d_in[0];
    const int*   eidx = (const int*)d_in[1];
    const float* Wk   = (const float*)d_in[2];
    const float* bk   = (const float*)d_in[3];
    const float* Wq   = (const float*)d_in[4];
    const float* bq   = (const float*)d_in[5];
    const float* Wv   = (const float*)d_in[6];
    const float* bv   = (const float*)d_in[7];
    const float* Wa   = (const float*)d_in[8];
    const float* ba   = (const float*)d_in[9];
    const float* skip = (const float*)d_in[10];
    const float* arel = (const float*)d_in[11];
    const float* mrel = (const float*)d_in[12];
    const float* prel = (const float*)d_in[13];
    const float* lng  = (const float*)d_in[14];
    const float* lnb  = (const float*)d_in[15];

    const size_t nodeElems = (size_t)NTYPES * NNODES * CDIM;   // 12.8M floats
    float* b0 = (float*)d_ws;
    float* b1 = b0 + nodeElems;
    float* b2 = b1 + nodeElems;
    float* b3 = b2 + nodeElems;
    float* b4 = b3 + nodeElems;
    float* wlog = b4 + nodeElems;                     // E*H
    float* amax = wlog + (size_t)NEDGES * HDIM;       // T*N*H
    float* den  = amax + (size_t)NTYPES * NNODES * HDIM;

    float* kb = b0; float* qb = b1; float* vb = b2; float* agg = b3;
    const int metaS[2] = {0, 1}, metaD[2] = {1, 0};
    float* xbufs[2] = { b4, b1 };   // layer-output ping-pong (b1 dead by LN time)

    const dim3 blk(256);
    const int gemmGrid = (NNODES + 127) / 128;
    const int edgeGrid = (NEDGES + 7) / 8;
    const int flatGrid = (NEDGES * HDIM + 255) / 256;
    const int lnGrid   = (NTYPES * NNODES + 7) / 8;

    const float* xin = x0;
    for (int l = 0; l < NLAYERS; ++l) {
        // K/Q/V projections per node type (WMMA GEMMs)
        for (int t = 0; t < NTYPES; ++t) {
            const float* xt = xin + (size_t)t * NNODES * CDIM;
            const size_t wOff = ((size_t)l * NTYPES + t) * CDIM * CDIM;
            const size_t bOff = ((size_t)l * NTYPES + t) * CDIM;
            float* kt = kb + (size_t)t * NNODES * CDIM;
            float* qt = qb + (size_t)t * NNODES * CDIM;
            float* vt = vb + (size_t)t * NNODES * CDIM;
            gemm128_bias<<<gemmGrid, blk, 0, stream>>>(xt, Wk + wOff, bk + bOff, kt, NNODES, 0);
            gemm128_bias<<<gemmGrid, blk, 0, stream>>>(xt, Wq + wOff, bq + bOff, qt, NNODES, 0);
            gemm128_bias<<<gemmGrid, blk, 0, stream>>>(xt, Wv + wOff, bv + bOff, vt, NNODES, 0);
        }
        // init softmax scratch + aggregation buffer
        fill_f32<<<(int)((nodeElems + 255) / 256), blk, 0, stream>>>(agg, 0.0f, nodeElems);
        const size_t nhElems = (size_t)NTYPES * NNODES * HDIM;
        fill_f32<<<(int)((nhElems + 255) / 256), blk, 0, stream>>>(amax, -3.0e38f, nhElems);
        fill_f32<<<(int)((nhElems + 255) / 256), blk, 0, stream>>>(den, 0.0f, nhElems);

        // per-relation edge passes
        for (int e = 0; e < 2; ++e) {
            const int st = metaS[e], dt = metaD[e];
            const int* esrc = eidx + ((size_t)e * 2 + 0) * NEDGES;
            const int* edst = eidx + ((size_t)e * 2 + 1) * NEDGES;
            const float* ar = arel + ((size_t)l * 2 + e) * HDIM * DDIM * DDIM;
            const float* mr = mrel + ((size_t)l * 2 + e) * HDIM * DDIM * DDIM;
            const float* pr = prel + ((size_t)l * 2 + e) * HDIM;
            const float* kS = kb + (size_t)st * NNODES * CDIM;
            const float* qD = qb + (size_t)dt * NNODES * CDIM;
            const float* vS = vb + (size_t)st * NNODES * CDIM;
            float* amaxD = amax + (size_t)dt * NNODES * HDIM;
            float* denD  = den  + (size_t)dt * NNODES * HDIM;
            float* aggD  = agg  + (size_t)dt * NNODES * CDIM;

            edge_logits<<<edgeGrid, blk, 0, stream>>>(kS, qD, esrc, edst, ar, pr,
                                                      wlog, amaxD, NEDGES);
            edge_exp<<<flatGrid, blk, 0, stream>>>(wlog, edst, amaxD, denD, NEDGES);
            edge_agg<<<edgeGrid, blk, 0, stream>>>(vS, esrc, edst, mr, wlog, denD,
                                                   aggD, NEDGES);
        }

        // o = gelu(agg) @ Wa + ba   (WMMA GEMM with fused GELU) ; reuse kb as o
        float* ob = kb;
        for (int t = 0; t < NTYPES; ++t) {
            const size_t wOff = ((size_t)l * NTYPES + t) * CDIM * CDIM;
            const size_t bOff = ((size_t)l * NTYPES + t) * CDIM;
            gemm128_bias<<<gemmGrid, blk, 0, stream>>>(
                agg + (size_t)t * NNODES * CDIM, Wa + wOff, ba + bOff,
                ob + (size_t)t * NNODES * CDIM, NNODES, 1);
        }

        // skip + LayerNorm + ReLU
        float* xout = xbufs[l];
        skip_ln_relu<<<lnGrid, blk, 0, stream>>>(
            ob, xin, skip + (size_t)l * NTYPES,
            lng + (size_t)l * NTYPES * CDIM, lnb + (size_t)l * NTYPES * CDIM,
            xout, NTYPES * NNODES);
        xin = xout;
    }

    // output = final x[type 0]  (first NNODES rows of the last buffer)
    hipMemcpyAsync(d_out, xin, (size_t)NNODES * CDIM * sizeof(float),
                   hipMemcpyDeviceToDevice, stream);
}